// AMG_encoder_8160437862446
// MI455X (gfx1250) — compile-verified
//
#include <hip/hip_runtime.h>
#include <hip/hip_bf16.h>

// AMG multi-grid trilinear encoder for gfx1250 (MI455X).
// - Repack feature grids [64][2][64^3] -> [64][64^3][2] (float2) in d_ws so each
//   corner is ONE global_load_b64 (halves L2 gather ops; 128MB fits MI455X 192MB L2).
// - Main kernel: 256 threads = 8 wave32; each wave handles 8 grids for 32 points.
//   Point coords + per-grid affine params staged into LDS with
//   global_load_async_to_lds_b128 (gfx1250 async-to-LDS path, ASYNCcnt).
// - Results staged in padded LDS tile [32][132], then streamed out as one
//   contiguous 16KB block with global_store_async_from_lds_b128 (coalesced).

#define N_GRIDS   64
#define N_FEATS   2
#define GDIM      64
#define VOXELS    262144        // 64^3
#define BATCH     1048576
#define TILE_PTS  32
#define OUT_CH    128           // N_GRIDS * N_FEATS
#define TILE_PAD  132           // 132*4 = 528 bytes, 16B aligned rows, bank-spread

__device__ __forceinline__ int iclamp63(int v) {
    return v < 0 ? 0 : (v > 63 ? 63 : v);
}

__global__ __launch_bounds__(256) void amg_repack(const float* __restrict__ fg,
                                                  float2* __restrict__ pk) {
    size_t v = (size_t)blockIdx.x * 256 + threadIdx.x;   // 0 .. 64*262144-1
    int g = (int)(v >> 18);
    int i = (int)(v & (VOXELS - 1));
    const float* gp = fg + ((size_t)g << 19);            // g * 2 * VOXELS
    float a = gp[i];
    float b = gp[VOXELS + i];
    pk[v] = make_float2(a, b);
}

template <bool PACKED>
__global__ __launch_bounds__(256) void amg_main(const float*  __restrict__ x,
                                                const float*  __restrict__ fg,
                                                const float2* __restrict__ pk,
                                                const float*  __restrict__ scales,
                                                const float*  __restrict__ trans,
                                                float*        __restrict__ out) {
    __shared__ __attribute__((aligned(16))) float s_x[TILE_PTS * 3];     // 384 B
    __shared__ __attribute__((aligned(16))) float s_scale[N_GRIDS * 3];  // 768 B
    __shared__ __attribute__((aligned(16))) float s_trans[N_GRIDS * 3];  // 768 B
    __shared__ __attribute__((aligned(16))) float s_tile[TILE_PTS][TILE_PAD]; // ~16.5 KB

    const int t    = threadIdx.x;
    const int lane = t & 31;
    const int wave = t >> 5;
    const long long p0 = (long long)blockIdx.x * TILE_PTS;

    // ---- async stage: x tile (24 x b128), scales (48 x b128), trans (48 x b128)
    if (t < 24) {
        const float* ga = x + p0 * 3 + t * 4;
        unsigned la = (unsigned)(unsigned long long)&s_x[t * 4];
        asm volatile("global_load_async_to_lds_b128 %0, %1, off"
                     :: "v"(la), "v"(ga) : "memory");
    } else if (t >= 32 && t < 80) {
        int j = t - 32;
        const float* ga = scales + j * 4;
        unsigned la = (unsigned)(unsigned long long)&s_scale[j * 4];
        asm volatile("global_load_async_to_lds_b128 %0, %1, off"
                     :: "v"(la), "v"(ga) : "memory");
    } else if (t >= 96 && t < 144) {
        int j = t - 96;
        const float* ga = trans + j * 4;
        unsigned la = (unsigned)(unsigned long long)&s_trans[j * 4];
        asm volatile("global_load_async_to_lds_b128 %0, %1, off"
                     :: "v"(la), "v"(ga) : "memory");
    }
    asm volatile("s_wait_asynccnt 0" ::: "memory");
    __syncthreads();

    const float px = s_x[lane * 3 + 0];
    const float py = s_x[lane * 3 + 1];
    const float pz = s_x[lane * 3 + 2];
    const float INV = 1.0f / 1.48f;
    const float HALF = 31.5f; // 0.5 * (64 - 1)

#pragma unroll
    for (int i = 0; i < 8; ++i) {
        const int g = (wave << 3) + i;   // uniform per wave
        const float sx = s_scale[g * 3 + 0], sy = s_scale[g * 3 + 1], sz = s_scale[g * 3 + 2];
        const float tx = s_trans[g * 3 + 0], ty = s_trans[g * 3 + 1], tz = s_trans[g * 3 + 2];

        const float ix = (fmaf(px, sx, tx) * INV + 1.0f) * HALF;
        const float iy = (fmaf(py, sy, ty) * INV + 1.0f) * HALF;
        const float iz = (fmaf(pz, sz, tz) * INV + 1.0f) * HALF;

        const float fxf = floorf(ix), fyf = floorf(iy), fzf = floorf(iz);
        const int x0 = (int)fxf, y0 = (int)fyf, z0 = (int)fzf;
        const float fx = ix - fxf, fy = iy - fyf, fz = iz - fzf;

        // per-corner validity folded into 1-D weights (zeros padding)
        const float wx0 = ((unsigned)x0       <= 63u) ? (1.0f - fx) : 0.0f;
        const float wx1 = ((unsigned)(x0 + 1) <= 63u) ? fx          : 0.0f;
        const float wy0 = ((unsigned)y0       <= 63u) ? (1.0f - fy) : 0.0f;
        const float wy1 = ((unsigned)(y0 + 1) <= 63u) ? fy          : 0.0f;
        const float wz0 = ((unsigned)z0       <= 63u) ? (1.0f - fz) : 0.0f;
        const float wz1 = ((unsigned)(z0 + 1) <= 63u) ? fz          : 0.0f;

        const int cx0 = iclamp63(x0), cx1 = iclamp63(x0 + 1);
        const int cy0 = iclamp63(y0), cy1 = iclamp63(y0 + 1);
        const int cz0 = iclamp63(z0), cz1 = iclamp63(z0 + 1);

        const int zy00 = (cz0 << 12) | (cy0 << 6);
        const int zy10 = (cz0 << 12) | (cy1 << 6);
        const int zy01 = (cz1 << 12) | (cy0 << 6);
        const int zy11 = (cz1 << 12) | (cy1 << 6);

        float2 v000, v100, v010, v110, v001, v101, v011, v111;
        if (PACKED) {
            const float2* gp = pk + ((size_t)g << 18);
            v000 = gp[zy00 + cx0]; v100 = gp[zy00 + cx1];
            v010 = gp[zy10 + cx0]; v110 = gp[zy10 + cx1];
            v001 = gp[zy01 + cx0]; v101 = gp[zy01 + cx1];
            v011 = gp[zy11 + cx0]; v111 = gp[zy11 + cx1];
        } else {
            const float* g0 = fg + ((size_t)g << 19);
            const float* g1 = g0 + VOXELS;
            v000 = make_float2(g0[zy00 + cx0], g1[zy00 + cx0]);
            v100 = make_float2(g0[zy00 + cx1], g1[zy00 + cx1]);
            v010 = make_float2(g0[zy10 + cx0], g1[zy10 + cx0]);
            v110 = make_float2(g0[zy10 + cx1], g1[zy10 + cx1]);
            v001 = make_float2(g0[zy01 + cx0], g1[zy01 + cx0]);
            v101 = make_float2(g0[zy01 + cx1], g1[zy01 + cx1]);
            v011 = make_float2(g0[zy11 + cx0], g1[zy11 + cx0]);
            v111 = make_float2(g0[zy11 + cx1], g1[zy11 + cx1]);
        }

        const float w00 = wy0 * wz0, w10 = wy1 * wz0;
        const float w01 = wy0 * wz1, w11 = wy1 * wz1;
        const float c000 = wx0 * w00, c100 = wx1 * w00;
        const float c010 = wx0 * w10, c110 = wx1 * w10;
        const float c001 = wx0 * w01, c101 = wx1 * w01;
        const float c011 = wx0 * w11, c111 = wx1 * w11;

        float a0 = v000.x * c000;
        a0 = fmaf(v100.x, c100, a0); a0 = fmaf(v010.x, c010, a0);
        a0 = fmaf(v110.x, c110, a0); a0 = fmaf(v001.x, c001, a0);
        a0 = fmaf(v101.x, c101, a0); a0 = fmaf(v011.x, c011, a0);
        a0 = fmaf(v111.x, c111, a0);
        float a1 = v000.y * c000;
        a1 = fmaf(v100.y, c100, a1); a1 = fmaf(v010.y, c010, a1);
        a1 = fmaf(v110.y, c110, a1); a1 = fmaf(v001.y, c001, a1);
        a1 = fmaf(v101.y, c101, a1); a1 = fmaf(v011.y, c011, a1);
        a1 = fmaf(v111.y, c111, a1);

        *(float2*)&s_tile[lane][2 * g] = make_float2(a0, a1);  // ds_store_b64
    }

    __syncthreads();

    // ---- coalesced async store of the 32x128 output tile (16 KB contiguous)
    const int r = t >> 3;             // row 0..31
    const int cbase = (t & 7) * 16;   // 8 threads per 128-col row
    float* grow = out + (p0 + r) * OUT_CH + cbase;
#pragma unroll
    for (int j = 0; j < 4; ++j) {
        unsigned la = (unsigned)(unsigned long long)&s_tile[r][cbase + j * 4];
        const float* ga = grow + j * 4;
        asm volatile("global_store_async_from_lds_b128 %0, %1, off"
                     :: "v"(ga), "v"(la) : "memory");
    }
    asm volatile("s_wait_asynccnt 0" ::: "memory");
}

extern "C" void kernel_launch(void* const* d_in, const int* in_sizes, int n_in,
                              void* d_out, int out_size, void* d_ws, size_t ws_size,
                              hipStream_t stream) {
    const float* x      = (const float*)d_in[0];
    const float* fg     = (const float*)d_in[1];
    const float* scales = (const float*)d_in[2];
    const float* trans  = (const float*)d_in[3];
    float* out          = (float*)d_out;

    const size_t pk_bytes = (size_t)N_GRIDS * VOXELS * sizeof(float2); // 128 MB
    const int n_tiles = BATCH / TILE_PTS;                              // 32768

    if (ws_size >= pk_bytes) {
        float2* pk = (float2*)d_ws;
        amg_repack<<<(N_GRIDS * VOXELS) / 256, 256, 0, stream>>>(fg, pk);
        amg_main<true><<<n_tiles, 256, 0, stream>>>(x, fg, pk, scales, trans, out);
    } else {
        amg_main<false><<<n_tiles, 256, 0, stream>>>(x, fg, nullptr, scales, trans, out);
    }
}